// PointTransformerLayer_53944789238361
// MI455X (gfx1250) — compile-verified
//
#include <hip/hip_runtime.h>
#include <hip/hip_bf16.h>

typedef float v2f __attribute__((ext_vector_type(2)));
typedef float v8f __attribute__((ext_vector_type(8)));

#define C 64
#define S 8
#define CS 8   // C/S

// ---------------------------------------------------------------------------
// Zero the output accumulator (harness poisons d_out with 0xAA).
// ---------------------------------------------------------------------------
__global__ void pt_zero_kernel(float* __restrict__ out, int n) {
    int i = blockIdx.x * blockDim.x + threadIdx.x;
    if (i < n) out[i] = 0.0f;
}

// ---------------------------------------------------------------------------
// Stage 1: Y = X @ W + b   (X: Nrows x 64, W: 64 x 64, row-major, fp32)
// One wave32 per 16-row tile; K=64 via 16 chained V_WMMA_F32_16X16X4_F32,
// 4 column tiles of 16 held in 4 v8f accumulators.
//
// f32 16x16x4 WMMA VGPR layouts (ISA 7.12.2):
//   A (16x4):  lane l (l<16): row=l, holds K=0,1 in v0,v1; lanes 16-31: K=2,3
//   B (4x16):  lane l: col=l%16, v0/v1 hold K = (l/16)*2 + {0,1}
//   C/D 16x16: vgpr r on lane l = D[r + 8*(l/16)][l%16]
// ---------------------------------------------------------------------------
__global__ void pt_proj_kernel(const float* __restrict__ X,
                               const float* __restrict__ W,
                               const float* __restrict__ bias,
                               float* __restrict__ Y,
                               int Nrows) {
    const int lane = threadIdx.x & 31;
    const int warp = threadIdx.x >> 5;
    const int tile = blockIdx.x * (blockDim.x >> 5) + warp;
    if (tile * 16 >= Nrows) return;

    const int rlo   = lane & 15;          // row (A) / col (B,D) within tile
    const int khalf = (lane >> 4) * 2;    // K sub-offset for this half-wave
    const int mhigh = (lane >> 4) * 8;    // M offset for C/D rows

    // clamp row for (theoretical) ragged tail; N=50000 is a multiple of 16
    int arow_idx = tile * 16 + rlo;
    if (arow_idx >= Nrows) arow_idx = Nrows - 1;
    const float* __restrict__ arow = X + (size_t)arow_idx * C;

    v8f acc0 = {}, acc1 = {}, acc2 = {}, acc3 = {};

    for (int k0 = 0; k0 < C; k0 += 4) {
        v2f a;
        a.x = arow[k0 + khalf + 0];
        a.y = arow[k0 + khalf + 1];

        const float* __restrict__ w0 = W + (size_t)(k0 + khalf) * C;
        v2f b;

        b.x = w0[0 * 16 + rlo];       b.y = w0[C + 0 * 16 + rlo];
        acc0 = __builtin_amdgcn_wmma_f32_16x16x4_f32(false, a, false, b,
                                                     (short)0, acc0, false, false);
        b.x = w0[1 * 16 + rlo];       b.y = w0[C + 1 * 16 + rlo];
        acc1 = __builtin_amdgcn_wmma_f32_16x16x4_f32(false, a, false, b,
                                                     (short)0, acc1, false, false);
        b.x = w0[2 * 16 + rlo];       b.y = w0[C + 2 * 16 + rlo];
        acc2 = __builtin_amdgcn_wmma_f32_16x16x4_f32(false, a, false, b,
                                                     (short)0, acc2, false, false);
        b.x = w0[3 * 16 + rlo];       b.y = w0[C + 3 * 16 + rlo];
        acc3 = __builtin_amdgcn_wmma_f32_16x16x4_f32(false, a, false, b,
                                                     (short)0, acc3, false, false);
    }

    // store D + bias
    v8f accs[4] = {acc0, acc1, acc2, acc3};
#pragma unroll
    for (int nt = 0; nt < 4; ++nt) {
        const int col = nt * 16 + rlo;
        const float bv = bias[col];
#pragma unroll
        for (int r = 0; r < 8; ++r) {
            int m = tile * 16 + mhigh + r;
            if (m < Nrows) Y[(size_t)m * C + col] = accs[nt][r] + bv;
        }
    }
}

// ---------------------------------------------------------------------------
// Stage 2: per-edge kernel. 8 lanes per edge; lane owns channel group
// s -> channels [8s, 8s+8). Gathers are contiguous 32B per lane (256B/edge).
// ---------------------------------------------------------------------------
__global__ void pt_edge_kernel(const float* __restrict__ edges,
                               const int*   __restrict__ eidx,
                               const float* __restrict__ xq,
                               const float* __restrict__ xk,
                               const float* __restrict__ xv,
                               const float* __restrict__ Wp,
                               const float* __restrict__ bp,
                               const float* __restrict__ W1,
                               const float* __restrict__ b1,
                               const float* __restrict__ W2,
                               const float* __restrict__ b2,
                               float* __restrict__ out,
                               int E) {
    __shared__ float sW1[C * CS];   // 64 x 8
    __shared__ float sW2[CS * CS];  // 8 x 8
    __shared__ float sWp[16];
    __shared__ float sb1[CS];
    __shared__ float sb2[CS];
    __shared__ float sbp;

    for (int i = threadIdx.x; i < C * CS; i += blockDim.x) sW1[i] = W1[i];
    if (threadIdx.x < CS * CS) sW2[threadIdx.x] = W2[threadIdx.x];
    if (threadIdx.x < 16)      sWp[threadIdx.x] = Wp[threadIdx.x];
    if (threadIdx.x < CS)      { sb1[threadIdx.x] = b1[threadIdx.x];
                                 sb2[threadIdx.x] = b2[threadIdx.x]; }
    if (threadIdx.x == 0)      sbp = bp[0];
    __syncthreads();

    const int gtid     = blockIdx.x * blockDim.x + threadIdx.x;
    const int s        = gtid & 7;                       // channel group
    const int e_stride = (gridDim.x * blockDim.x) >> 3;

    for (int e = gtid >> 3; e < E; e += e_stride) {
        const int src = eidx[2 * e + 0];
        const int dst = eidx[2 * e + 1];

        // p_r = edges[e] . Wp + bp   (2 elements per lane, reduce over 8)
        const float* erow = edges + (size_t)e * 16;
        float pr = erow[2 * s] * sWp[2 * s] + erow[2 * s + 1] * sWp[2 * s + 1];
        pr += __shfl_xor(pr, 1, 8);
        pr += __shfl_xor(pr, 2, 8);
        pr += __shfl_xor(pr, 4, 8);
        pr += sbp;

        // gather this lane's 8 channels of x_q[src], x_k[dst], x_v[dst]
        const float4* q4 = (const float4*)(xq + (size_t)src * C + s * CS);
        const float4* k4 = (const float4*)(xk + (size_t)dst * C + s * CS);
        const float4* v4 = (const float4*)(xv + (size_t)dst * C + s * CS);
        float4 qa = q4[0], qb = q4[1];
        float4 ka = k4[0], kb = k4[1];
        float4 va = v4[0], vb = v4[1];

        float t[CS];
        t[0] = fmaxf(ka.x - qa.x + pr, 0.0f);
        t[1] = fmaxf(ka.y - qa.y + pr, 0.0f);
        t[2] = fmaxf(ka.z - qa.z + pr, 0.0f);
        t[3] = fmaxf(ka.w - qa.w + pr, 0.0f);
        t[4] = fmaxf(kb.x - qb.x + pr, 0.0f);
        t[5] = fmaxf(kb.y - qb.y + pr, 0.0f);
        t[6] = fmaxf(kb.z - qb.z + pr, 0.0f);
        t[7] = fmaxf(kb.w - qb.w + pr, 0.0f);

        // h = relu(w) @ W1 : lane-partial over its 8 channels, reduce over 8
        float h[CS];
#pragma unroll
        for (int j = 0; j < CS; ++j) {
            float acc = 0.0f;
#pragma unroll
            for (int i = 0; i < CS; ++i)
                acc += t[i] * sW1[(s * CS + i) * CS + j];
            acc += __shfl_xor(acc, 1, 8);
            acc += __shfl_xor(acc, 2, 8);
            acc += __shfl_xor(acc, 4, 8);
            h[j] = fmaxf(acc + sb1[j], 0.0f);   // + b1, relu
        }

        // h2 = relu(h) @ W2 + b2   (redundant across the 8 lanes, tiny)
        float h2[CS];
#pragma unroll
        for (int j = 0; j < CS; ++j) {
            float acc = sb2[j];
#pragma unroll
            for (int i = 0; i < CS; ++i)
                acc += h[i] * sW2[i * CS + j];
            h2[j] = acc;
        }

        // softmax over the 8 logits
        float m = h2[0];
#pragma unroll
        for (int j = 1; j < CS; ++j) m = fmaxf(m, h2[j]);
        float wsm[CS];
        float denom = 0.0f;
#pragma unroll
        for (int j = 0; j < CS; ++j) { wsm[j] = __expf(h2[j] - m); denom += wsm[j]; }
        const float inv = 1.0f / denom;

        // msg[c=8s+i] = (x_v[c] + p_r) * w[i]; segment-sum by src via L2 atomics
        float* obase = out + (size_t)src * C + s * CS;
        float vv[CS] = {va.x, va.y, va.z, va.w, vb.x, vb.y, vb.z, vb.w};
#pragma unroll
        for (int i = 0; i < CS; ++i) {
            float msg = (vv[i] + pr) * (wsm[i] * inv);
            __hip_atomic_fetch_add(&obase[i], msg,
                                   __ATOMIC_RELAXED, __HIP_MEMORY_SCOPE_AGENT);
        }
    }
}

// ---------------------------------------------------------------------------
// Launcher
// ---------------------------------------------------------------------------
extern "C" void kernel_launch(void* const* d_in, const int* in_sizes, int n_in,
                              void* d_out, int out_size, void* d_ws, size_t ws_size,
                              hipStream_t stream) {
    const float* q     = (const float*)d_in[0];
    const float* k     = (const float*)d_in[1];
    const float* v     = (const float*)d_in[2];
    const float* edges = (const float*)d_in[3];
    const int*   eidx  = (const int*)  d_in[4];
    const float* Wq    = (const float*)d_in[5];
    const float* bq    = (const float*)d_in[6];
    const float* Wk    = (const float*)d_in[7];
    const float* bk    = (const float*)d_in[8];
    const float* Wv    = (const float*)d_in[9];
    const float* bv    = (const float*)d_in[10];
    const float* Wp    = (const float*)d_in[11];
    const float* bp    = (const float*)d_in[12];
    const float* W1    = (const float*)d_in[13];
    const float* b1    = (const float*)d_in[14];
    const float* W2    = (const float*)d_in[15];
    const float* b2    = (const float*)d_in[16];

    const int N = in_sizes[0] / C;   // 50000
    const int E = in_sizes[3] / 16;  // 800000

    float* xq = (float*)d_ws;
    float* xk = xq + (size_t)N * C;
    float* xv = xk + (size_t)N * C;

    // zero output accumulator
    {
        int n = out_size;
        pt_zero_kernel<<<(n + 255) / 256, 256, 0, stream>>>((float*)d_out, n);
    }

    // stage 1: three WMMA projections into workspace
    {
        const int tiles  = (N + 15) / 16;
        const int blocks = (tiles + 7) / 8;   // 8 waves / block
        pt_proj_kernel<<<blocks, 256, 0, stream>>>(q, Wq, bq, xq, N);
        pt_proj_kernel<<<blocks, 256, 0, stream>>>(k, Wk, bk, xk, N);
        pt_proj_kernel<<<blocks, 256, 0, stream>>>(v, Wv, bv, xv, N);
    }

    // stage 2: per-edge attention + scatter-add
    {
        const long long threads = (long long)E * 8;
        int blocks = (int)((threads + 255) / 256);
        pt_edge_kernel<<<blocks, 256, 0, stream>>>(edges, eidx, xq, xk, xv,
                                                   Wp, bp, W1, b1, W2, b2,
                                                   (float*)d_out, E);
    }
}